// Attention_3058016715095
// MI455X (gfx1250) — compile-verified
//
#include <hip/hip_runtime.h>
#include <hip/hip_bf16.h>
#include <stdint.h>

// Problem constants (from reference): B=32, S=2048, D=1024
#define BB 32
#define SS 2048
#define DD 1024
#define SCHUNKS 16              // S split for the two column-reduction passes

typedef __attribute__((ext_vector_type(2))) float v2f;
typedef __attribute__((ext_vector_type(8))) float v8f;

// ---------------------------------------------------------------------------
// Pass 1 / 6 helper: partial column sums of E over a chunk of S.
// grid = (B, SCHUNKS), block = 256 threads; thread t owns float4 #t of the
// D=1024 row (256*4 = 1024). Coalesced global_load_b128 stream over E.
// ---------------------------------------------------------------------------
__global__ void colsum_partial(const float* __restrict__ E,
                               float* __restrict__ partial) {
    const int b = blockIdx.x;
    const int chunk = blockIdx.y;
    const int t = threadIdx.x;
    const float4* Eb = (const float4*)(E + (size_t)b * SS * DD);
    float4 acc = make_float4(0.f, 0.f, 0.f, 0.f);
    const int s0 = chunk * (SS / SCHUNKS);
#pragma unroll 4
    for (int s = s0; s < s0 + SS / SCHUNKS; ++s) {
        float4 e = Eb[(size_t)s * (DD / 4) + t];
        acc.x += e.x; acc.y += e.y; acc.z += e.z; acc.w += e.w;
    }
    float4* P = (float4*)(partial + (size_t)(b * SCHUNKS + chunk) * DD);
    P[t] = acc;
}

// Weighted variant for zs = sum_s f[b,s] * E[b,s,:]
__global__ void wsum_partial(const float* __restrict__ E,
                             const float* __restrict__ fw,
                             float* __restrict__ partial) {
    const int b = blockIdx.x;
    const int chunk = blockIdx.y;
    const int t = threadIdx.x;
    const float4* Eb = (const float4*)(E + (size_t)b * SS * DD);
    float4 acc = make_float4(0.f, 0.f, 0.f, 0.f);
    const int s0 = chunk * (SS / SCHUNKS);
#pragma unroll 4
    for (int s = s0; s < s0 + SS / SCHUNKS; ++s) {
        const float w = fw[(size_t)b * SS + s];
        float4 e = Eb[(size_t)s * (DD / 4) + t];
        acc.x += w * e.x; acc.y += w * e.y; acc.z += w * e.z; acc.w += w * e.w;
    }
    float4* P = (float4*)(partial + (size_t)(b * SCHUNKS + chunk) * DD);
    P[t] = acc;
}

// Reduce the SCHUNKS partials and scale. grid = B, block = 256.
__global__ void reduce16_kernel(const float* __restrict__ partial,
                                float* __restrict__ out, float scale) {
    const int b = blockIdx.x;
    const int t = threadIdx.x;
    const float4* P = (const float4*)(partial + (size_t)b * SCHUNKS * DD);
    float4 acc = make_float4(0.f, 0.f, 0.f, 0.f);
#pragma unroll
    for (int c = 0; c < SCHUNKS; ++c) {
        float4 p = P[(size_t)c * (DD / 4) + t];
        acc.x += p.x; acc.y += p.y; acc.z += p.z; acc.w += p.w;
    }
    float4* O = (float4*)(out + (size_t)b * DD);
    O[t] = make_float4(acc.x * scale, acc.y * scale, acc.z * scale, acc.w * scale);
}

// ---------------------------------------------------------------------------
// V = Ys @ W^T via V_WMMA_F32_16X16X4_F32 (fp32 WMMA; matches fp32 data).
//   C[m=b, n=d] = sum_k Ys[b,k] * W[d,k]     (W row-major [d][e])
// One wave (block of 32) per 16x16 output tile; M tiles = B/16 = 2,
// N tiles = D/16 = 64 -> 128 blocks; K loop = 1024/4 = 256 WMMAs.
// A 16x4 layout: lane m=lane&15; VGPR0/1 = K {2*hi, 2*hi+1}, hi=lane>>4.
// B 4x16 mirrored: lane n=lane&15; VGPR0/1 = K {2*hi, 2*hi+1}.
// D layout: acc[r] -> (M = r + 8*hi, N = lane&15).
// ---------------------------------------------------------------------------
__global__ void gemm_v_wmma(const float* __restrict__ ys,
                            const float* __restrict__ w,
                            float* __restrict__ v) {
    const int tile = blockIdx.x;
    const int tm = tile & 1;        // M (batch) tile: 0..1
    const int tn = tile >> 1;       // N (d) tile:    0..63
    const int lane = threadIdx.x;   // 0..31, full wave (EXEC all 1s)
    const int hi = lane >> 4;
    const int lo = lane & 15;

    const int m = tm * 16 + lo;     // A row (batch index)
    const int n = tn * 16 + lo;     // B col (d index)
    const float2* ys2 = (const float2*)ys;  // row strides are even
    const float2* w2 = (const float2*)w;

    v8f acc = {};
    for (int k0 = 0; k0 < DD; k0 += 4) {
        const int ka = k0 + 2 * hi;                  // even
        float2 af = ys2[((size_t)m * DD + ka) >> 1]; // Ys[m, ka..ka+1]
        float2 bf = w2[((size_t)n * DD + ka) >> 1];  // W[n, ka..ka+1]
        v2f a; a.x = af.x; a.y = af.y;
        v2f b; b.x = bf.x; b.y = bf.y;
        acc = __builtin_amdgcn_wmma_f32_16x16x4_f32(
            false, a, false, b, (short)0, acc, false, false);
    }
#pragma unroll
    for (int r = 0; r < 8; ++r) {
        const int mo = tm * 16 + r + 8 * hi;   // batch
        const int no = tn * 16 + lo;           // d
        v[(size_t)mo * DD + no] = acc[r];
    }
}

// ---------------------------------------------------------------------------
// f_raw[b,s] = mask ? E[b,s,:] . v[b,:] : -1e9
// grid = (B, S/8), block = 256 (8 waves, one row per wave).
// v[b] cached in LDS (4 KB); lane does 8 float4 MACs then wave32 shuffle tree.
// ---------------------------------------------------------------------------
__global__ void dot_kernel(const float* __restrict__ E,
                           const float* __restrict__ v,
                           const unsigned char* __restrict__ mask,
                           float* __restrict__ f_raw) {
    __shared__ float vsh[DD];
    const int b = blockIdx.x;
    for (int i = threadIdx.x; i < DD; i += blockDim.x) vsh[i] = v[(size_t)b * DD + i];
    __syncthreads();

    const int wave = threadIdx.x >> 5;
    const int lane = threadIdx.x & 31;
    const int s = blockIdx.y * 8 + wave;

    const float4* row = (const float4*)(E + ((size_t)b * SS + s) * DD);
    const float4* v4 = (const float4*)vsh;
    float acc = 0.f;
#pragma unroll
    for (int i = 0; i < DD / 4 / 32; ++i) {         // 8 iterations
        float4 e = row[lane + i * 32];
        float4 vv = v4[lane + i * 32];
        acc += e.x * vv.x + e.y * vv.y + e.z * vv.z + e.w * vv.w;
    }
#pragma unroll
    for (int off = 16; off > 0; off >>= 1) acc += __shfl_xor(acc, off, 32);

    if (lane == 0) {
        const size_t idx = (size_t)b * SS + s;
        f_raw[idx] = mask[idx] ? acc : -1.0e9f;
    }
}

// ---------------------------------------------------------------------------
// Per-batch softmax over S=2048. grid = B, block = 256.
// ---------------------------------------------------------------------------
__global__ void softmax_kernel(const float* __restrict__ f_raw,
                               float* __restrict__ f_out) {
    __shared__ float red[256];
    const int b = blockIdx.x;
    const int t = threadIdx.x;
    const float* fr = f_raw + (size_t)b * SS;

    float m = -3.4e38f;
    for (int s = t; s < SS; s += 256) m = fmaxf(m, fr[s]);
    red[t] = m; __syncthreads();
    for (int off = 128; off > 0; off >>= 1) {
        if (t < off) red[t] = fmaxf(red[t], red[t + off]);
        __syncthreads();
    }
    m = red[0]; __syncthreads();

    float sum = 0.f;
    for (int s = t; s < SS; s += 256) sum += __expf(fr[s] - m);
    red[t] = sum; __syncthreads();
    for (int off = 128; off > 0; off >>= 1) {
        if (t < off) red[t] += red[t + off];
        __syncthreads();
    }
    const float inv = 1.0f / red[0];

    for (int s = t; s < SS; s += 256)
        f_out[(size_t)b * SS + s] = __expf(fr[s] - m) * inv;
}

// ---------------------------------------------------------------------------
// Launch: zs = d_out[0 : B*D], f = d_out[B*D : B*D + B*S]
// Scratch layout (floats): partial[B*16*D] | ys[B*D] | v[B*D] | f_raw[B*S]
// ---------------------------------------------------------------------------
extern "C" void kernel_launch(void* const* d_in, const int* in_sizes, int n_in,
                              void* d_out, int out_size, void* d_ws, size_t ws_size,
                              hipStream_t stream) {
    const float* E = (const float*)d_in[0];
    const unsigned char* mask = (const unsigned char*)d_in[1];  // jnp bool -> 1 byte
    const float* W = (const float*)d_in[2];

    float* out = (float*)d_out;
    float* zs = out;                 // [B, D]
    float* f = out + BB * DD;        // [B, S]

    float* ws = (float*)d_ws;
    float* partial = ws;                              // B*16*D = 524288 floats
    float* ys = partial + (size_t)BB * SCHUNKS * DD;  // B*D
    float* v = ys + (size_t)BB * DD;                  // B*D
    float* f_raw = v + (size_t)BB * DD;               // B*S

    const dim3 gridCS(BB, SCHUNKS);

    // 1) ys = mean_s E  (stream E once)
    colsum_partial<<<gridCS, 256, 0, stream>>>(E, partial);
    reduce16_kernel<<<BB, 256, 0, stream>>>(partial, ys, 1.0f / (float)SS);

    // 2) v[b] = W @ ys[b] as a 32x1024x1024 GEMM on fp32 WMMA
    gemm_v_wmma<<<(BB / 16) * (DD / 16), 32, 0, stream>>>(ys, W, v);

    // 3) f_raw[b,s] = mask ? E[b,s].v[b] : -1e9   (stream E second time)
    dot_kernel<<<dim3(BB, SS / 8), 256, 0, stream>>>(E, v, mask, f_raw);

    // 4) softmax over S -> f (directly into d_out)
    softmax_kernel<<<BB, 256, 0, stream>>>(f_raw, f);

    // 5) zs = sum_s f * E   (stream E third time)
    wsum_partial<<<gridCS, 256, 0, stream>>>(E, f, partial);
    reduce16_kernel<<<BB, 256, 0, stream>>>(partial, zs, 1.0f);
}